// Loss_9423158247904
// MI455X (gfx1250) — compile-verified
//
#include <hip/hip_runtime.h>
#include <hip/hip_bf16.h>
#include <stdint.h>

// Problem constants (fixed by the reference setup)
#define B_DIM      1024
#define C_DIM      100000
#define THREADS    256
#define TILE_BYTES 16384          // 4096 floats per tile
#define NPAIRS     12             // 24 tiles = 12 pairs; 24*4096 = 98304 floats
#define REM_F4     424            // (100000 - 98304) / 4 remainder float4s

// exp(64*x) == exp2((64*log2(e)) * x) ; one v_mul + one v_exp_f32 per element
#define K_EXP2  92.33248261689366f
#define S_SCALE 64.0f
#define MARGIN  0.35f

// Async DMA global->LDS, 16B/lane. ISA: the 24-bit immediate offset is added
// to BOTH the LDS address (vdst) and the global address (vaddr). Since our
// global tile stride (16KB) equals the LDS buffer stride, a single fixed vdst
// and a vaddr bumped by 32KB per tile-PAIR lets the immediate select both the
// global tile and the correct half of the LDS double buffer at once.
template<int OFF>
__device__ __forceinline__ void async_ld_b128(uint32_t lds, uint64_t ga) {
    asm volatile("global_load_async_to_lds_b128 %0, %1, off offset:%2 th:TH_LOAD_NT"
                 :: "v"(lds), "v"(ga), "i"(OFF)
                 : "memory");
}

template<int BASE>  // BASE = 0 (even tile -> buf half 0) or 16384 (odd -> half 1)
__device__ __forceinline__ void issue_tile(uint32_t lds, uint64_t ga) {
    async_ld_b128<BASE +     0>(lds, ga);
    async_ld_b128<BASE +  4096>(lds, ga);
    async_ld_b128<BASE +  8192>(lds, ga);
    async_ld_b128<BASE + 12288>(lds, ga);
}

__device__ __forceinline__ void wait_async_le4() {
    asm volatile("s_wait_asynccnt 0x4" ::: "memory");
}
__device__ __forceinline__ void wait_async_0() {
    asm volatile("s_wait_asynccnt 0x0" ::: "memory");
}

__global__ __launch_bounds__(THREADS)
void row_loss_kernel(const float* __restrict__ x,
                     const int*   __restrict__ label,
                     float*       __restrict__ row_L) {
    __shared__ __align__(16) float buf[2 * 4096];   // contiguous 32 KB double buffer
    __shared__ float wave_part[THREADS / 32];

    const int row = blockIdx.x;
    const int tid = threadIdx.x;
    const float* __restrict__ xrow = x + (size_t)row * C_DIM;

    // Fixed per-thread LDS target (low 32 bits of flat shared ptr = LDS byte addr)
    const uint32_t lds = (uint32_t)(uintptr_t)(&buf[0]) + (uint32_t)(tid * 16);
    // Rolling per-thread global base; advanced 32 KB once per tile pair
    uint64_t ga = (uint64_t)(uintptr_t)xrow + (uint64_t)(tid * 16);

    float a0 = 0.0f, a1 = 0.0f, a2 = 0.0f, a3 = 0.0f;

    // Consume this wave's own async-loaded slices (per-wave ASYNCcnt sync only)
    auto consume_half = [&](int half) {
        const float4* p = (const float4*)(&buf[half * 4096]);
#pragma unroll
        for (int i = 0; i < 4; ++i) {
            float4 v = p[i * THREADS + tid];
            a0 += __builtin_amdgcn_exp2f(K_EXP2 * v.x);
            a1 += __builtin_amdgcn_exp2f(K_EXP2 * v.y);
            a2 += __builtin_amdgcn_exp2f(K_EXP2 * v.z);
            a3 += __builtin_amdgcn_exp2f(K_EXP2 * v.w);
        }
    };

    issue_tile<0>(lds, ga);                    // tile 0 -> buf half 0
#pragma unroll 1
    for (int p = 0; p < NPAIRS; ++p) {
        issue_tile<TILE_BYTES>(lds, ga);       // tile 2p+1 -> buf half 1
        wait_async_le4();                      // tile 2p landed
        consume_half(0);
        const uint64_t ga_next = ga + 2 * TILE_BYTES;
        if (p + 1 < NPAIRS) {
            issue_tile<0>(lds, ga_next);       // tile 2p+2 -> buf half 0
            wait_async_le4();                  // tile 2p+1 landed
        } else {
            wait_async_0();                    // drain
        }
        consume_half(1);
        ga = ga_next;
    }

    // Remainder: 424 float4s, plain coalesced b128 loads
    {
        const float4* xrem = (const float4*)(xrow + (size_t)NPAIRS * 2 * 4096);
        for (int i = tid; i < REM_F4; i += THREADS) {
            float4 v = xrem[i];
            a0 += __builtin_amdgcn_exp2f(K_EXP2 * v.x);
            a1 += __builtin_amdgcn_exp2f(K_EXP2 * v.y);
            a2 += __builtin_amdgcn_exp2f(K_EXP2 * v.z);
            a3 += __builtin_amdgcn_exp2f(K_EXP2 * v.w);
        }
    }

    float acc = (a0 + a1) + (a2 + a3);

    // Wave32 reduction, then cross-wave via LDS
#pragma unroll
    for (int off = 16; off > 0; off >>= 1)
        acc += __shfl_xor(acc, off, 32);
    if ((tid & 31) == 0) wave_part[tid >> 5] = acc;
    __syncthreads();

    if (tid == 0) {
        float rowsum = 0.0f;
#pragma unroll
        for (int w = 0; w < THREADS / 32; ++w) rowsum += wave_part[w];
        const int   y   = label[row];
        const float xy  = xrow[y];
        const float num = S_SCALE * (xy - MARGIN);
        const float den = __expf(num) + (rowsum - __expf(S_SCALE * xy));
        row_L[row] = (num - __logf(den)) / S_SCALE;
    }
}

__global__ __launch_bounds__(THREADS)
void finalize_kernel(const float* __restrict__ row_L, float* __restrict__ out) {
    __shared__ float wave_part[THREADS / 32];
    const int tid = threadIdx.x;
    float a = 0.0f;
    for (int i = tid; i < B_DIM; i += THREADS) a += row_L[i];
#pragma unroll
    for (int off = 16; off > 0; off >>= 1)
        a += __shfl_xor(a, off, 32);
    if ((tid & 31) == 0) wave_part[tid >> 5] = a;
    __syncthreads();
    if (tid == 0) {
        float s = 0.0f;
#pragma unroll
        for (int w = 0; w < THREADS / 32; ++w) s += wave_part[w];
        out[0] = -s / (float)B_DIM;
    }
}

extern "C" void kernel_launch(void* const* d_in, const int* in_sizes, int n_in,
                              void* d_out, int out_size, void* d_ws, size_t ws_size,
                              hipStream_t stream) {
    const float* x     = (const float*)d_in[0];
    const int*   label = (const int*)d_in[1];
    float*       rowL  = (float*)d_ws;      // 1024 floats of scratch
    float*       out   = (float*)d_out;

    row_loss_kernel<<<B_DIM, THREADS, 0, stream>>>(x, label, rowL);
    finalize_kernel<<<1, THREADS, 0, stream>>>(rowL, out);
}